// VoxelMorph_71296457114147
// MI455X (gfx1250) — compile-verified
//
#include <hip/hip_runtime.h>

// ---------------------------------------------------------------------------
// VoxelMorph loss (NCC 9^3 box-filter + flow-grad + temporal smoothness)
// for MI455X / gfx1250 (wave32, WMMA, async-to-LDS).
//
// NCC is fused: per (16x16 HxW tile, 32-deep D segment, pair) block we stream
// D-slices through LDS:
//   phase1: GLOBAL_LOAD_ASYNC_TO_LDS_B32 of 24x24 halo slice of I and J
//           (zero-padding via once-zeroed staging + EXEC-masked OOB lanes)
//   phase2: 5 products + 9-tap W-filter                          [VALU]
//   phase3: 9-tap H-filter as 6x V_WMMA_F32_16X16X4_F32 (banded  [WMMA]
//           ones A-matrix, exact f32), scatter into 9-slice ring
//           buffer + running D-window sums
//   phase4: emit cc for output slice d-4, accumulate
// Traffic ~141MB vs ~530MB for a 3-pass global implementation; the whole
// problem is HBM-bound (~0.26GB total -> ~11us at 23.3 TB/s). WMMA keeps the
// filter arithmetic off the VALU in exact f32; async-to-LDS removes the
// global->VGPR->LDS staging hop.
// ---------------------------------------------------------------------------

#define DHW   128
#define PLANE (DHW * DHW)          // 16384
#define VOL   (DHW * DHW * DHW)    // 2097152
#define WIN   9
#define PADR  4
#define TILE  16
#define REGN  24                   // TILE + 2*PADR
#define SEGD  32
#define NSEG  (DHW / SEGD)         // 4
#define NPAIR 3

typedef __attribute__((ext_vector_type(2))) float v2f;
typedef __attribute__((ext_vector_type(8))) float v8f;

#if defined(__gfx1250__) && __has_builtin(__builtin_amdgcn_global_load_async_to_lds_b32)
#define HAVE_ASYNC_LDS 1
typedef __attribute__((address_space(1))) int as1_int;   // global (AS1)
typedef __attribute__((address_space(3))) int as3_int;   // LDS (AS3)
#else
#define HAVE_ASYNC_LDS 0
#endif

__global__ __launch_bounds__(256)
void ncc_kernel(const float* __restrict__ image,      // frame 0 used
                const float* __restrict__ pred_image, // frames 1..3 used
                float* __restrict__ cc_accum)
{
    const int tid  = threadIdx.x;
    const int lane = tid & 31;
    const int wave = tid >> 5;

    const int tileW = (blockIdx.x & 7) * TILE;
    const int tileH = (blockIdx.x >> 3) * TILE;
    const int seg   = blockIdx.y;
    const int pair  = blockIdx.z;

    const float* __restrict__ Ibase = image;                              // t = 0
    const float* __restrict__ Jbase = pred_image + (size_t)(pair + 1) * VOL;

    // ---- LDS (total 64,164 B) ----
    __shared__ float regI[REGN][REGN + 1];        // 24x25 staging, +1 pad
    __shared__ float regJ[REGN][REGN + 1];
    __shared__ float S1[5][REGN][TILE + 1];       // W-filtered, 5 fields, 24x17
    __shared__ float ring[9][5][TILE][TILE];      // 9-slice D ring buffer
    __shared__ float Rsum[5][TILE][TILE];         // running 9-slice sums
    __shared__ float redAcc;

    {   // zero ring + running sums + staging (staging zeros realize the
        // H/W zero-padding: OOB cells are never overwritten by the loads)
        float* rp = &ring[0][0][0][0];
        for (int i = tid; i < 9 * 5 * TILE * TILE; i += 256) rp[i] = 0.f;
        float* sp = &Rsum[0][0][0];
        for (int i = tid; i < 5 * TILE * TILE; i += 256) sp[i] = 0.f;
        for (int i = tid; i < REGN * (REGN + 1); i += 256) {
            (&regI[0][0])[i] = 0.f;
            (&regJ[0][0])[i] = 0.f;
        }
        if (tid == 0) redAcc = 0.f;
    }
    __syncthreads();   // DScnt flushed: zeros land before any async LDS write

    const int dOut0 = seg * SEGD;
    const int dOut1 = dOut0 + SEGD;
    float ccAcc = 0.f;
    int   slot  = 8;                                  // wraps to 0 first iter

    for (int dIn = dOut0 - PADR; dIn < dOut1 + PADR; ++dIn) {
        slot = (slot == 8) ? 0 : slot + 1;            // ring slot, no div
        const bool inVol = (dIn >= 0) && (dIn < DHW);

        // -------- phase 1: stage 24x24 halo slice into LDS ------------------
        if (inVol) {
            const size_t dOff = (size_t)dIn * PLANE;
            for (int idx = tid; idx < REGN * REGN; idx += 256) {
                const int rh = idx / REGN, rx = idx - rh * REGN;
                const int hh = tileH - PADR + rh;
                const int ww = tileW - PADR + rx;
                if ((unsigned)hh < (unsigned)DHW && (unsigned)ww < (unsigned)DHW) {
                    const size_t off = dOff + (size_t)hh * DHW + (size_t)ww;
#if HAVE_ASYNC_LDS
                    // direct global->LDS, ASYNCcnt-tracked, no VGPR hop
                    __builtin_amdgcn_global_load_async_to_lds_b32(
                        (as1_int*)(unsigned long long)(Ibase + off),
                        (as3_int*)(unsigned long long)(&regI[rh][rx]), 0, 0);
                    __builtin_amdgcn_global_load_async_to_lds_b32(
                        (as1_int*)(unsigned long long)(Jbase + off),
                        (as3_int*)(unsigned long long)(&regJ[rh][rx]), 0, 0);
#else
                    regI[rh][rx] = Ibase[off];
                    regJ[rh][rx] = Jbase[off];
#endif
                    if (dIn + 1 < DHW) {              // next-slice prefetch
                        __builtin_prefetch(Ibase + off + PLANE, 0, 1);
                        __builtin_prefetch(Jbase + off + PLANE, 0, 1);
                    }
                }
#if !HAVE_ASYNC_LDS
                else {
                    regI[rh][rx] = 0.f;
                    regJ[rh][rx] = 0.f;
                }
#endif
            }
        }
#if HAVE_ASYNC_LDS
        asm volatile("s_wait_asynccnt 0x0" ::: "memory");
#endif
        __syncthreads();

        // -------- phase 2: 5 products + 9-tap W filter -> S1[f][24][16] -----
        if (inVol) {
            for (int idx = tid; idx < REGN * TILE; idx += 256) {
                const int h = idx >> 4, wo = idx & 15;
                float sI = 0.f, sJ = 0.f, sII = 0.f, sJJ = 0.f, sIJ = 0.f;
#pragma unroll
                for (int k = 0; k < WIN; ++k) {
                    const float a = regI[h][wo + k];
                    const float b = regJ[h][wo + k];
                    sI += a; sJ += b; sII += a * a; sJJ += b * b; sIJ += a * b;
                }
                S1[0][h][wo] = sI;  S1[1][h][wo] = sJ;
                S1[2][h][wo] = sII; S1[3][h][wo] = sJJ;
                S1[4][h][wo] = sIJ;
            }
        }
        __syncthreads();

        // -------- phase 3: 9-tap H filter via V_WMMA_F32_16X16X4_F32 --------
        // D[i][n] = sum_{j=i..i+8} S1[f][j][n], decomposed into 6 K=4 chunks
        // with banded-ones A matrices (exact f32 accumulation).
        if (wave < 5) {
            const int f  = wave;
            const int n  = lane & 15;                 // B/D column (w)
            const int i  = lane & 15;                 // A row (h_out)
            const int kb = (lane < 16) ? 0 : 2;       // VGPR0 holds K0/K2
            v8f acc = {};
            if (inVol) {
#pragma unroll
                for (int c = 0; c < 6; ++c) {
                    const int j0 = 4 * c + kb;
                    const int j1 = j0 + 1;
                    v2f a, b;
                    a.x = (j0 >= i && j0 <= i + 8) ? 1.f : 0.f;
                    a.y = (j1 >= i && j1 <= i + 8) ? 1.f : 0.f;
                    b.x = S1[f][j0][n];
                    b.y = S1[f][j1][n];
                    acc = __builtin_amdgcn_wmma_f32_16x16x4_f32(
                        /*neg_a=*/false, a, /*neg_b=*/false, b,
                        /*c_mod=*/(short)0, acc,
                        /*reuse_a=*/false, /*reuse_b=*/false);
                }
            }
            // D layout: VGPR r holds M=r (lanes 0-15) / M=r+8 (lanes 16-31)
            const int hb = (lane < 16) ? 0 : 8;
#pragma unroll
            for (int r = 0; r < 8; ++r) {
                const int   h  = r + hb;
                const float nv = acc[r];
                const float ov = ring[slot][f][h][n];
                ring[slot][f][h][n] = nv;
                Rsum[f][h][n] += nv - ov;             // running D-window sum
            }
        }
        __syncthreads();

        // -------- phase 4: emit cc for output slice dIn - 4 -----------------
        const int dO = dIn - PADR;
        if (dO >= dOut0 && dO < dOut1) {
            const int h = tid >> 4, w = tid & 15;     // 256 threads = 16x16
            const float Is  = Rsum[0][h][w];
            const float Js  = Rsum[1][h][w];
            const float IIs = Rsum[2][h][w];
            const float JJs = Rsum[3][h][w];
            const float IJs = Rsum[4][h][w];
            const float inv = 1.f / 729.f;            // 1 / WIN^3
            const float cross = IJs - Is * Js * inv;
            const float Iv    = IIs - Is * Is * inv;
            const float Jv    = JJs - Js * Js * inv;
            ccAcc += cross * cross / (Iv * Jv + 1e-5f);
        }
        __syncthreads();   // protect S1/reg/ring/Rsum across iterations
    }

    atomicAdd(&redAcc, ccAcc);                        // ds_add_f32
    __syncthreads();
    if (tid == 0) atomicAdd(cc_accum, redAcc);
}

// ---- flow smoothness: sum of squared forward diffs over 3 axes ------------
__global__ __launch_bounds__(256)
void grad_kernel(const float* __restrict__ flow, float* __restrict__ accum,
                 int total)
{
    __shared__ float red;
    if (threadIdx.x == 0) red = 0.f;
    __syncthreads();
    float s = 0.f;
    const int stride = gridDim.x * blockDim.x;
    for (int idx = blockIdx.x * blockDim.x + threadIdx.x; idx < total; idx += stride) {
        const float v   = flow[idx];
        const int   rem = idx & (VOL - 1);
        const int   w   = rem & 127;
        const int   h   = (rem >> 7) & 127;
        const int   d   = rem >> 14;
        if (w < 127) { const float t = flow[idx + 1]     - v; s += t * t; }
        if (h < 127) { const float t = flow[idx + DHW]   - v; s += t * t; }
        if (d < 127) { const float t = flow[idx + PLANE] - v; s += t * t; }
    }
    atomicAdd(&red, s);
    __syncthreads();
    if (threadIdx.x == 0) atomicAdd(accum, red);
}

// ---- per-frame spatial sum of pred_mask -----------------------------------
__global__ __launch_bounds__(256)
void area_kernel(const float* __restrict__ pm, float* __restrict__ areas)
{
    const int t = blockIdx.y;
    const float* __restrict__ p = pm + (size_t)t * VOL;
    __shared__ float red;
    if (threadIdx.x == 0) red = 0.f;
    __syncthreads();
    float s = 0.f;
    const int stride = gridDim.x * blockDim.x;
    for (int idx = blockIdx.x * blockDim.x + threadIdx.x; idx < VOL; idx += stride)
        s += p[idx];
    atomicAdd(&red, s);
    __syncthreads();
    if (threadIdx.x == 0) atomicAdd(&areas[t], red);
}

// ---- combine everything into the scalar loss ------------------------------
__global__ void finalize_kernel(const float* __restrict__ ws,
                                const float* __restrict__ td,
                                float* __restrict__ out)
{
    if (threadIdx.x != 0 || blockIdx.x != 0) return;
    const float ncc  = 1.f - ws[0] / (3.f * (float)VOL);
    const float gN   = 3.f * 9.f * 127.f * 128.f * 128.f;  // 3 axes, same count
    const float grad = ws[1] / gN;

    float a[4];
    float amax = -1e30f;
    for (int t = 0; t < 4; ++t) { a[t] = ws[2 + t]; amax = fmaxf(amax, a[t]); }
    for (int t = 0; t < 4; ++t) a[t] /= amax;

    float sc = 0.f; int cnt = 0;
    for (int i = 0; i < 2; ++i)
        for (int j = i + 1; j < 3; ++j)
            for (int k = j + 1; k < 4; ++k) {
                const float lam = (td[j] - td[i] + 1e-5f) / (td[k] - td[i] + 1e-5f);
                const float v = a[j] - a[i] - lam * (a[k] - a[i]);
                sc += v * v; ++cnt;
            }
    sc /= (float)cnt;
    out[0] = ncc + 0.1f * grad + 0.1f * sc;
}

__global__ void init_ws(float* ws)
{
    if (threadIdx.x < 6) ws[threadIdx.x] = 0.f;
}

extern "C" void kernel_launch(void* const* d_in, const int* in_sizes, int n_in,
                              void* d_out, int out_size, void* d_ws, size_t ws_size,
                              hipStream_t stream)
{
    const float* image      = (const float*)d_in[0];
    // d_in[1] = mask (unused by the reference loss)
    const float* pred_image = (const float*)d_in[2];
    const float* pred_mask  = (const float*)d_in[3];
    const float* flow       = (const float*)d_in[4];
    const float* time_diff  = (const float*)d_in[5];
    float* ws  = (float*)d_ws;   // [0]=cc sum, [1]=grad sum, [2..5]=areas
    float* out = (float*)d_out;

    init_ws<<<1, 32, 0, stream>>>(ws);

    dim3 gNcc(64, NSEG, NPAIR);                      // 8x8 HW tiles, 4 D segs, 3 pairs
    ncc_kernel<<<gNcc, 256, 0, stream>>>(image, pred_image, &ws[0]);

    grad_kernel<<<1024, 256, 0, stream>>>(flow, &ws[1], 9 * VOL);

    dim3 gArea(256, 4, 1);
    area_kernel<<<gArea, 256, 0, stream>>>(pred_mask, &ws[2]);

    finalize_kernel<<<1, 32, 0, stream>>>(ws, time_diff, out);
}